// Decoder_26225070309401
// MI455X (gfx1250) — compile-verified
//
#include <hip/hip_runtime.h>
#include <hip/hip_bf16.h>
#include <math.h>

// ---------------------------------------------------------------------------
// Pointer-network decoder for MI455X (gfx1250).
// One persistent workgroup (128 threads = 4 wave32) per batch row:
//   preamble : h_bar / query0 / mask0 / first_idx, and
//              keys[b] = high_node[b] @ Wk via V_WMMA_F32_16X16X4_F32 -> LDS
//              (each wave owns 2 N-tile columns; A fragment loaded once per
//               k-step as a single b64 and shared by 2 WMMAs; row guard done
//               with clamped addresses + v_cndmask, no EXEC branching)
//   loop i=0..99 : q matvec, scores vs LDS keys, tanh/clip/mask,
//                  log-softmax, threefry2x32 Gumbel categorical (JAX-exact),
//                  mask/reward/query-carry updates.
// ---------------------------------------------------------------------------

#define B_TOT 2048
#define LMAX 100
#define EDIM 128
#define KSTR 129          // LDS row stride (129 mod 64 == 1 -> conflict-free-ish)
#define NEG_BIG (-100000000.0f)
#define CLIPC 10.0f

typedef __attribute__((ext_vector_type(2))) float v2f;
typedef __attribute__((ext_vector_type(8))) float v8f;

__device__ __forceinline__ unsigned rotl32(unsigned x, unsigned r) {
  return (x << r) | (x >> (32u - r));
}

// JAX threefry2x32 (20 rounds), identical key schedule/rotations.
__device__ __forceinline__ void threefry2x32(unsigned k0, unsigned k1,
                                             unsigned x0, unsigned x1,
                                             unsigned &o0, unsigned &o1) {
  const unsigned ks2 = 0x1BD11BDAu ^ k0 ^ k1;
  unsigned y0 = x0 + k0, y1 = x1 + k1;
#define TF_R4(a, b, c, d)                                                      \
  y0 += y1; y1 = rotl32(y1, a); y1 ^= y0;                                      \
  y0 += y1; y1 = rotl32(y1, b); y1 ^= y0;                                      \
  y0 += y1; y1 = rotl32(y1, c); y1 ^= y0;                                      \
  y0 += y1; y1 = rotl32(y1, d); y1 ^= y0;
  TF_R4(13, 15, 26, 6)  y0 += k1;  y1 += ks2 + 1u;
  TF_R4(17, 29, 16, 24) y0 += ks2; y1 += k0 + 2u;
  TF_R4(13, 15, 26, 6)  y0 += k0;  y1 += k1 + 3u;
  TF_R4(17, 29, 16, 24) y0 += k1;  y1 += ks2 + 4u;
  TF_R4(13, 15, 26, 6)  y0 += ks2; y1 += k0 + 5u;
#undef TF_R4
  o0 = y0; o1 = y1;
}

// word f (0..199) of jax.random.split(key(123), 100) flat layout:
// counts=iota(200) split in halves; out = concat(o0,o1).reshape(100,2)
__device__ __forceinline__ unsigned split_keyword(unsigned f) {
  unsigned o0, o1;
  if (f < (unsigned)LMAX) {
    threefry2x32(0u, 123u, f, f + LMAX, o0, o1);
    return o0;
  } else {
    threefry2x32(0u, 123u, f - LMAX, f, o0, o1);
    return o1;
  }
}

// JAX uniform(tiny,1) -> Gumbel, from raw 32 random bits.
__device__ __forceinline__ float gumbel_from_bits(unsigned bits) {
  const float tiny = 1.17549435e-38f;
  unsigned fb = (bits >> 9) | 0x3f800000u;
  float f = __uint_as_float(fb) - 1.0f;
  float u = f * (1.0f - tiny) + tiny;
  u = fmaxf(tiny, u);
  return -logf(-logf(u));
}

__global__ __launch_bounds__(128) void decoder_kernel(
    const float *__restrict__ high_node, const float *__restrict__ original_node,
    const int *__restrict__ num_cell, const float *__restrict__ costs,
    const float *__restrict__ init_w, const float *__restrict__ Wc,
    const float *__restrict__ bc, const float *__restrict__ Wv,
    const float *__restrict__ bv, const float *__restrict__ Wq,
    const float *__restrict__ Wk, float *__restrict__ out_logprob,
    float *__restrict__ out_reward, float *__restrict__ out_action) {
  __shared__ float s_keys[LMAX * KSTR];   // 51.6 KB
  __shared__ float s_query[EDIM];
  __shared__ float s_q[EDIM];
  __shared__ float s_h[EDIM];
  __shared__ float s_inith[EDIM];
  __shared__ float s_hbar[EDIM];
  __shared__ float s_mean[EDIM];
  __shared__ float s_logits[128];
  __shared__ float s_red[128];
  __shared__ int s_redi[128];
  __shared__ int s_mask[LMAX];
  __shared__ int s_first;

  const int b = blockIdx.x;
  const int t = threadIdx.x;   // 0..127
  const int lane = t & 31;
  const int wave = t >> 5;
  const float *hn = high_node + (size_t)b * LMAX * EDIM;
  const float *onb = original_node + (size_t)b * LMAX * 4;
  const float *cb = costs + (size_t)b * LMAX;
  const int ncell = num_cell[b];
  const float SQRT_H = 11.313708498984761f;  // sqrt(128)

  // ------------------------- preamble -------------------------------------
  {
    float acc = 0.f;
    for (int l = 0; l < LMAX; ++l) acc += hn[l * EDIM + t];
    s_mean[t] = acc / (float)LMAX;
    if (t < LMAX) s_mask[t] = (hn[t * EDIM] == 0.0f) ? 1 : 0;
    if (t < 4) s_mask[t] = 1;
    if (t == 0) {
      int fi = 0;
      float cv = cb[0];
      for (int j = 1; j < 4; ++j)
        if (cb[j] < cv) { cv = cb[j]; fi = j; }
      s_first = fi;
    }
  }
  __syncthreads();
  {
    float hb = bc[t];
    for (int k = 0; k < EDIM; ++k) hb += s_mean[k] * Wc[k * EDIM + t];
    float q0 = bv[t];
    for (int k = 0; k < 2 * EDIM; ++k) q0 += init_w[k] * Wv[k * EDIM + t];
    s_hbar[t] = hb;
    s_query[t] = hb + q0;
    s_inith[t] = 0.f;
  }

  // ---------------- keys[b] = high_node[b] @ Wk via f32 WMMA ---------------
  // 7 M-tiles (rows, padded 100->112); each wave owns 2 of 8 N-tiles and
  // walks all M-tiles, so one A fragment (b64 load) feeds two WMMAs.
  {
    const int ml = lane & 15;
    const int h2 = lane >> 4;           // 0/1: K-half within fragment
    const int n0 = wave * 32;           // two 16-col N-tiles per wave
    const float *wk0 = Wk + n0 + ml;
    for (int mt = 0; mt < 7; ++mt) {
      const int l0 = mt * 16;
      const int row = l0 + ml;
      const bool rowok = row < LMAX;
      const int rowc = rowok ? row : (LMAX - 1);   // clamped: always in-bounds
      const float *arow = hn + rowc * EDIM + 2 * h2;
      v8f acc0 = {}, acc1 = {};
      for (int k0 = 0; k0 < EDIM; k0 += 4) {
        // unconditional aligned b64 load, zero via cndmask (no EXEC branches)
        const float2 av = *(const float2 *)(arow + k0);
        v2f a;
        a.x = rowok ? av.x : 0.0f;
        a.y = rowok ? av.y : 0.0f;
        const int ka = k0 + 2 * h2;
        v2f b0, b1;
        b0.x = wk0[ka * EDIM];
        b0.y = wk0[(ka + 1) * EDIM];
        b1.x = wk0[ka * EDIM + 16];
        b1.y = wk0[(ka + 1) * EDIM + 16];
        acc0 = __builtin_amdgcn_wmma_f32_16x16x4_f32(
            false, a, false, b0, (short)0, acc0, false, false);
        acc1 = __builtin_amdgcn_wmma_f32_16x16x4_f32(
            false, a, false, b1, (short)0, acc1, false, false);
      }
      const int rbase = l0 + (h2 ? 8 : 0);
#pragma unroll
      for (int j = 0; j < 8; ++j) {
        const int r = rbase + j;
        if (r < LMAX) {
          s_keys[r * KSTR + n0 + ml] = acc0[j];
          s_keys[r * KSTR + n0 + 16 + ml] = acc1[j];
        }
      }
    }
  }
  __syncthreads();

  // ------------------------- decode loop -----------------------------------
  float lp_sum = 0.f, rw_sum = 0.f;
  int st_idx = 0;

  for (int i = 0; i < LMAX; ++i) {
    // q = query @ Wq   (LDS broadcast reads, coalesced Wq rows from L2)
    float qv = 0.f;
    for (int k = 0; k < EDIM; ++k) qv += s_query[k] * Wq[k * EDIM + t];
    s_q[t] = qv;
    __syncthreads();

    // scores -> clipped tanh logits with mask
    float logit = -INFINITY;
    if (t < LMAX) {
      float sc = 0.f;
      const float *kr = &s_keys[t * KSTR];
      for (int k = 0; k < EDIM; ++k) sc += s_q[k] * kr[k];
      sc = sc / SQRT_H;
      float lg = CLIPC * tanhf(sc);
      logit = (s_mask[t] == 1) ? NEG_BIG : lg;
    }
    s_logits[t] = logit;

    // max reduction
    s_red[t] = logit;
    __syncthreads();
    for (int s = 64; s > 0; s >>= 1) {
      if (t < s) s_red[t] = fmaxf(s_red[t], s_red[t + s]);
      __syncthreads();
    }
    const float mx = s_red[0];
    __syncthreads();

    // sum-exp reduction -> logsumexp
    s_red[t] = (t < LMAX) ? expf(logit - mx) : 0.0f;
    __syncthreads();
    for (int s = 64; s > 0; s >>= 1) {
      if (t < s) s_red[t] += s_red[t + s];
      __syncthreads();
    }
    const float lse = mx + logf(s_red[0]);
    __syncthreads();

    // Gumbel categorical (JAX-exact threefry counters) + first-index argmax
    const unsigned sk0 = split_keyword((unsigned)(2 * i));
    const unsigned sk1 = split_keyword((unsigned)(2 * i + 1));
    float val = -INFINITY;
    if (t < LMAX) {
      const unsigned HALF = (unsigned)(B_TOT * LMAX) / 2u;
      const unsigned f = (unsigned)(b * LMAX + t);
      unsigned o0, o1, bits;
      if (f < HALF) { threefry2x32(sk0, sk1, f, f + HALF, o0, o1); bits = o0; }
      else          { threefry2x32(sk0, sk1, f - HALF, f, o0, o1); bits = o1; }
      val = logit + gumbel_from_bits(bits);
    }
    s_red[t] = val;
    s_redi[t] = t;
    __syncthreads();
    for (int s = 64; s > 0; s >>= 1) {
      if (t < s) {
        const float v2 = s_red[t + s];
        const int i2 = s_redi[t + s];
        if (v2 > s_red[t] || (v2 == s_red[t] && i2 < s_redi[t])) {
          s_red[t] = v2;
          s_redi[t] = i2;
        }
      }
      __syncthreads();
    }
    const int samp = s_redi[0];
    const int idx = (i == 0) ? s_first : samp;  // uniform across threads
    __syncthreads();

    // bookkeeping (thread 0): log-prob, reward, action
    if (t == 0) {
      const bool done = (ncell <= i);
      const float lp = s_logits[idx] - lse;
      lp_sum += done ? 0.0f : lp;
      const float sx = onb[st_idx * 4 + 2], sy = onb[st_idx * 4 + 3];
      const float ex = onb[idx * 4 + 0], ey = onb[idx * 4 + 1];
      const float dx = ex - sx, dy = ey - sy;
      const float ext = sqrtf(dx * dx + dy * dy);
      const float intr = cb[st_idx] + cb[idx];
      const float rw = (ext + intr) / 70.0f;
      rw_sum += (done || i == 0) ? 0.0f : rw;
      out_action[(size_t)b * LMAX + i] = (float)idx;  // int action stored as f32
    }
    st_idx = idx;

    // carry updates: mask group-of-4, h gather, new query
    const int base = (idx >> 2) << 2;
    if (t < 4) s_mask[base + t] = 1;
    const float hvt = hn[idx * EDIM + t];
    s_h[t] = hvt;
    if (i == 0) s_inith[t] = hvt;
    __syncthreads();

    float nq = bv[t] + s_hbar[t];
    for (int k = 0; k < EDIM; ++k) nq += s_inith[k] * Wv[k * EDIM + t];
    for (int k = 0; k < EDIM; ++k) nq += s_h[k] * Wv[(EDIM + k) * EDIM + t];
    s_query[t] = nq;
    __syncthreads();
  }

  if (t == 0) {
    out_logprob[b] = lp_sum;
    out_reward[b] = rw_sum;
  }
}

extern "C" void kernel_launch(void *const *d_in, const int *in_sizes, int n_in,
                              void *d_out, int out_size, void *d_ws,
                              size_t ws_size, hipStream_t stream) {
  (void)in_sizes; (void)n_in; (void)out_size; (void)d_ws; (void)ws_size;
  const float *high_node     = (const float *)d_in[0];
  const float *original_node = (const float *)d_in[1];
  // d_in[2] = map_in (unused by reference)
  const int   *num_cell      = (const int *)d_in[3];
  const float *costs         = (const float *)d_in[4];
  const float *init_w        = (const float *)d_in[5];
  const float *Wc            = (const float *)d_in[6];
  const float *bc            = (const float *)d_in[7];
  const float *Wv            = (const float *)d_in[8];
  const float *bv            = (const float *)d_in[9];
  const float *Wq            = (const float *)d_in[10];
  const float *Wk            = (const float *)d_in[11];

  float *out = (float *)d_out;
  float *out_logprob = out;                 // [0, 2048)
  float *out_reward  = out + B_TOT;         // [2048, 4096)
  float *out_action  = out + 2 * B_TOT;     // [4096, 4096 + 204800)

  decoder_kernel<<<dim3(B_TOT), dim3(128), 0, stream>>>(
      high_node, original_node, num_cell, costs, init_w, Wc, bc, Wv, bv, Wq, Wk,
      out_logprob, out_reward, out_action);
}